// SphericalGMMNet_57071525429465
// MI455X (gfx1250) — compile-verified
//
#include <hip/hip_runtime.h>
#include <math.h>
#include <stdint.h>

#define PIf 3.14159265358979323846f

typedef __attribute__((ext_vector_type(2))) float v2f;
typedef __attribute__((ext_vector_type(8))) float v8f;

// ---------------- complex helpers ----------------
__device__ __forceinline__ float2 cmul(float2 a, float2 b) {
  return make_float2(a.x * b.x - a.y * b.y, a.x * b.y + a.y * b.x);
}
__device__ __forceinline__ float2 cmulc(float2 a, float2 b) {  // a * conj(b)
  return make_float2(a.x * b.x + a.y * b.y, a.y * b.x - a.x * b.y);
}

// ---------------- WMMA: D = A(16x4) * B(4x16) + C (fp32) ----------------
__device__ __forceinline__ v8f wmma4(v2f a, v2f b, v8f c) {
  return __builtin_amdgcn_wmma_f32_16x16x4_f32(false, a, false, b, (short)0, c,
                                               false, false);
}

// Batched complex GEMM: C[z] = A[z](MxK) * B[z](KxN), row-major complex
// (float2). REQUIRES: M % 16 == 0, N % 16 == 0, K % 4 == 0 (true for all
// SO(3) layers: M = 16*P, N = P*O (O mult of 16), K = I*P (I mult of 4)).
// One wave computes one 16x16 tile; 8 waves per block.
// grid = (ceil(tilesN/8), tilesM, L), block = (32, 8)
__global__ void __launch_bounds__(256)
cgemm16_wmma(const float2* __restrict__ A, const float2* __restrict__ B,
             float2* __restrict__ C, int M, int N, int K, long long sA,
             long long sB, long long sC) {
  const float2* Ab = A + (long long)blockIdx.z * sA;
  const float2* Bb = B + (long long)blockIdx.z * sB;
  float2* Cb = C + (long long)blockIdx.z * sC;
  int row0 = blockIdx.y * 16;
  int col0 = (blockIdx.x * blockDim.y + threadIdx.y) * 16;
  if (col0 >= N) return;  // wave-uniform early-out (EXEC all-ones inside)
  int lane = threadIdx.x;
  int hh = lane >> 4;   // 0: holds K pair {0,1}; 1: holds K pair {2,3}
  int l16 = lane & 15;
  int arow = row0 + l16;
  int bcol = col0 + l16;
  // A fragment: complex pair (ka, ka+1) of one row = 16 contiguous bytes.
  const float4* aq = (const float4*)(Ab + (long long)arow * K) + hh;
  const float2* bq = Bb + bcol + (long long)hh * 2 * N;
  const long long bstep = 4LL * N;
  v8f rr = {}, ii = {}, ri = {}, ir = {};
  for (int k0 = 0; k0 < K; k0 += 4) {
    float4 av = aq[0];        // global_load_b128
    float2 b0 = bq[0];        // global_load_b64
    float2 b1 = bq[N];        // global_load_b64
    __builtin_prefetch(bq + bstep, 0, 1);  // speculative, next K-slab of B
    v2f ar = {av.x, av.z}, ai = {av.y, av.w};
    v2f br = {b0.x, b1.x}, bi = {b0.y, b1.y};
    rr = wmma4(ar, br, rr);
    ii = wmma4(ai, bi, ii);
    ri = wmma4(ar, bi, ri);
    ir = wmma4(ai, br, ir);
    aq += 2;
    bq += bstep;
  }
  for (int r = 0; r < 8; ++r) {
    int row = row0 + r + hh * 8;
    Cb[(long long)row * N + col0 + l16] =
        make_float2(rr[r] - ii[r], ri[r] + ir[r]);
  }
}

// ---------------- constant tables ----------------
// sc layout (floats): [0..152) quad betas per layer at qoff{0,64,96,116,132,144}
// [160..312) DH weights (same per-layer offsets, base 160)
// [320) A2[24], [344) B2[24], [368) A3[144], [512) B3G[144], [656) G3[144]
__global__ void k_smallconst(float* sc) {
  const int bw[6] = {32, 16, 10, 8, 6, 4};
  const int qoff[6] = {0, 64, 96, 116, 132, 144};
  int tid = threadIdx.x;
  int stride = blockDim.x;
  for (int li = 0; li < 6; ++li) {
    int b = bw[li], n = 2 * b;
    for (int k = tid; k < n; k += stride) {
      float theta = PIf * (2 * k + 1) / (4.0f * b);
      sc[qoff[li] + k] = theta;
      float s = 0.f;
      for (int j = 0; j < b; ++j) {
        float jj = 2.f * j + 1.f;
        s += sinf(jj * theta) / jj;
      }
      sc[160 + qoff[li] + k] = (2.0f / b) * sinf(theta) * s;
    }
  }
  for (int p = tid; p < 24; p += stride) {
    sc[320 + p] = 2.f * PIf * (float)(p / 3) / 8.f;
    sc[344 + p] = (PIf / 8.f) * (float)((p % 3) + 1) / 3.f;
  }
  for (int p = tid; p < 144; p += stride) {
    sc[368 + p] = 2.f * PIf * (float)(p / 18) / 8.f;
    sc[512 + p] = (PIf / 8.f) * (float)(((p / 6) % 3) + 1) / 3.f;
    sc[656 + p] = 2.f * PIf * (float)(p % 6) / 6.f;
  }
}

__device__ __forceinline__ double lf(int x) { return lgamma((double)x + 1.0); }

// Wigner-d table: out[l][mp+L-1][m+L-1][b], size L*(2L-1)^2*nb
__global__ void k_wigner(int L, int nb, const float* __restrict__ betas,
                         float* __restrict__ out) {
  int P = 2 * L - 1;
  long long total = (long long)L * P * P * nb;
  long long idx = (long long)blockIdx.x * blockDim.x + threadIdx.x;
  if (idx >= total) return;
  int bidx = (int)(idx % nb);
  long long t = idx / nb;
  int mcol = (int)(t % P); t /= P;
  int mrow = (int)(t % P);
  int l = (int)(t / P);
  int mp = mrow - (L - 1);
  int m = mcol - (L - 1);
  if (mp < -l || mp > l || m < -l || m > l) { out[idx] = 0.f; return; }
  double beta = (double)betas[bidx];
  double ch = cos(beta * 0.5), sh = sin(beta * 0.5);
  double pref = 0.5 * (lf(l + m) + lf(l - m) + lf(l + mp) + lf(l - mp));
  int s0 = (m - mp) > 0 ? (m - mp) : 0;
  int s1 = (l + m) < (l - mp) ? (l + m) : (l - mp);
  double v = 0.0;
  for (int s = s0; s <= s1; ++s) {
    double lg = pref - (lf(l + m - s) + lf(s) + lf(mp - m + s) + lf(l - mp - s));
    double term = exp(lg) * pow(ch, (double)(2 * l + m - mp - 2 * s)) *
                  pow(sh, (double)(mp - m + 2 * s));
    v += ((mp - m + s) & 1) ? -term : term;
  }
  out[idx] = (float)v;
}

// wc[l][m][n][p] = DK * exp(-i(fm*A3[p] + fn*G3[p]))
__global__ void k_wc_so3(int L, const float* __restrict__ dk,
                         const float* __restrict__ A3,
                         const float* __restrict__ G3,
                         float2* __restrict__ wc) {
  int P = 2 * L - 1;
  long long total = (long long)L * P * P * 144;
  long long idx = (long long)blockIdx.x * blockDim.x + threadIdx.x;
  if (idx >= total) return;
  int p = (int)(idx % 144);
  long long t = idx / 144;
  int n = (int)(t % P); t /= P;
  int m = (int)(t % P);
  float fm = (float)(m - (L - 1));
  float fn = (float)(n - (L - 1));
  float ang = -(fm * A3[p] + fn * G3[p]);
  float s, c;
  sincosf(ang, &s, &c);
  float d = dk[idx];
  wc[idx] = make_float2(d * c, d * s);
}

// wc0[l][m][p] = DK0[l][m][mid][p] * exp(-i*fm*A2[p])  (L=16, P=31, 24 p's)
__global__ void k_wc_s2(const float* __restrict__ dkfull,
                        const float* __restrict__ A2, float2* __restrict__ wc) {
  int idx = blockIdx.x * blockDim.x + threadIdx.x;
  if (idx >= 16 * 31 * 24) return;
  int p = idx % 24;
  int t = idx / 24;
  int m = t % 31;
  int l = t / 31;
  float fm = (float)(m - 15);
  float ang = -fm * A2[p];
  float s, c;
  sincosf(ang, &s, &c);
  float d = dkfull[(((l * 31 + m) * 31) + 15) * 24 + p];
  wc[idx] = make_float2(d * c, d * s);
}

// ---------------- S2 layer (conv 0) ----------------
// x (16,64,64) -> xf (16,64,31): partial DFT over last axis, freqs -15..15
__global__ void k_s2_dft(const float* __restrict__ x, float2* __restrict__ xf) {
  int idx = blockIdx.x * blockDim.x + threadIdx.x;
  if (idx >= 16 * 64 * 31) return;
  int m = idx % 31;
  int t = idx / 31;
  int y = t % 64;
  int b = t / 64;
  float fm = (float)(m - 15);
  const float* row = x + (b * 64 + y) * 64;
  float es, ec;
  sincosf(-2.f * PIf * fm / 64.f, &es, &ec);
  float2 e = make_float2(ec, es), w = make_float2(1.f, 0.f);
  float2 acc = make_float2(0.f, 0.f);
  for (int xx = 0; xx < 64; ++xx) {
    acc.x += row[xx] * w.x;
    acc.y += row[xx] * w.y;
    w = cmul(w, e);
  }
  xf[idx] = acc;
}

// X0[l][b][m] = sum_k din0[l][m][mid][k] * wq[k] * xf[b][k][m]
__global__ void k_s2_X(const float* __restrict__ din0,
                       const float* __restrict__ wq0,
                       const float2* __restrict__ xf,
                       float2* __restrict__ X0) {
  int idx = blockIdx.x * blockDim.x + threadIdx.x;
  if (idx >= 16 * 16 * 31) return;
  int m = idx % 31;
  int t = idx / 31;
  int b = t % 16;
  int l = t / 16;
  float2 acc = make_float2(0.f, 0.f);
  for (int k = 0; k < 64; ++k) {
    float w = din0[((l * 31 + m) * 31 + 15) * 64 + k] * wq0[k];
    float2 v = xf[(b * 64 + k) * 31 + m];
    acc.x += w * v.x;
    acc.y += w * v.y;
  }
  X0[idx] = acc;
}

// Y0[l][n][o] = sum_p wc0[l][n][p] * k1[o][p]
__global__ void k_s2_Y(const float2* __restrict__ wc0,
                       const float* __restrict__ k1, float2* __restrict__ Y0) {
  int idx = blockIdx.x * blockDim.x + threadIdx.x;
  if (idx >= 16 * 31 * 16) return;
  int o = idx % 16;
  int t = idx / 16;
  int n = t % 31;
  int l = t / 31;
  float2 acc = make_float2(0.f, 0.f);
  for (int p = 0; p < 24; ++p) {
    float kv = k1[o * 24 + p];
    float2 w = wc0[(l * 31 + n) * 24 + p];
    acc.x += w.x * kv;
    acc.y += w.y * kv;
  }
  Y0[idx] = acc;
}

// Z[l][(b,m)][(n,o)] = X0[l,b,m] * conj(Y0[l,n,o])  (P=31, O=16)
__global__ void k_s2_Z(const float2* __restrict__ X0,
                       const float2* __restrict__ Y0, float2* __restrict__ Z) {
  long long total = 16LL * 16 * 31 * 31 * 16;
  long long idx = (long long)blockIdx.x * blockDim.x + threadIdx.x;
  if (idx >= total) return;
  int o = (int)(idx % 16);
  long long t = idx / 16;
  int n = (int)(t % 31); t /= 31;
  int m = (int)(t % 31); t /= 31;
  int b = (int)(t % 16);
  int l = (int)(t / 16);
  float2 xv = X0[(l * 16 + b) * 31 + m];
  float2 yv = Y0[(l * 31 + n) * 16 + o];
  Z[(long long)((l * 16 + b) * 31 + m) * (31 * 16) + n * 16 + o] = cmulc(xv, yv);
}

// ---------------- shared SO(3) iFFT (used by all 5 conv layers) ----------------
// Fm[b][kk][m][n][o] = sum_l (2l+1) * dout[l][m][n][kk] * Z[l][(b,m)][(n,o)]
__global__ void k_fm(const float2* __restrict__ Z, const float* __restrict__ dout,
                     int L, int P, int O, int n2, float2* __restrict__ Fm) {
  const int B = 16;
  long long total = (long long)B * n2 * P * P * O;
  long long idx = (long long)blockIdx.x * blockDim.x + threadIdx.x;
  if (idx >= total) return;
  int o = (int)(idx % O);
  long long t = idx / O;
  int n = (int)(t % P); t /= P;
  int m = (int)(t % P); t /= P;
  int kk = (int)(t % n2);
  int b = (int)(t / n2);
  float2 acc = make_float2(0.f, 0.f);
  for (int l = 0; l < L; ++l) {
    float w = (float)(2 * l + 1) * dout[((l * P + m) * P + n) * n2 + kk];
    float2 z = Z[(long long)((l * B + b) * P + m) * (P * O) + n * O + o];
    acc.x += w * z.x;
    acc.y += w * z.y;
  }
  Fm[(long long)(((b * n2 + kk) * P + m) * P + n) * O + o] = acc;
}

// G1[b][kk][m][o][x] = sum_n Fm[b][kk][m][n][o] * exp(+2pi i * fn * x / n2)
__global__ void k_idft1(const float2* __restrict__ Fm, int L, int P, int O,
                        int n2, float2* __restrict__ G1) {
  const int B = 16;
  long long total = (long long)B * n2 * P * O * n2;
  long long idx = (long long)blockIdx.x * blockDim.x + threadIdx.x;
  if (idx >= total) return;
  int x = (int)(idx % n2);
  long long t = idx / n2;
  int o = (int)(t % O); t /= O;
  int m = (int)(t % P); t /= P;
  int kk = (int)(t % n2);
  int b = (int)(t / n2);
  float step = 2.f * PIf * (float)x / (float)n2;
  float s0, c0, s1, c1;
  sincosf(-step * (float)(L - 1), &s0, &c0);  // start at fn = -(L-1)
  sincosf(step, &s1, &c1);
  float2 w = make_float2(c0, s0), e = make_float2(c1, s1);
  float2 acc = make_float2(0.f, 0.f);
  for (int n = 0; n < P; ++n) {
    float2 f = Fm[(long long)(((b * n2 + kk) * P + m) * P + n) * O + o];
    acc.x += f.x * w.x - f.y * w.y;
    acc.y += f.x * w.y + f.y * w.x;
    w = cmul(w, e);
  }
  G1[(long long)(((b * n2 + kk) * P + m) * O + o) * n2 + x] = acc;
}

// out[b][o][kk][y][x] = Re sum_m G1[b][kk][m][o][x] * exp(+2pi i * fm * y / n2)
__global__ void k_idft2(const float2* __restrict__ G1, int L, int P, int O,
                        int n2, float* __restrict__ out) {
  const int B = 16;
  long long total = (long long)B * O * n2 * n2 * n2;
  long long idx = (long long)blockIdx.x * blockDim.x + threadIdx.x;
  if (idx >= total) return;
  int x = (int)(idx % n2);
  long long t = idx / n2;
  int y = (int)(t % n2); t /= n2;
  int kk = (int)(t % n2); t /= n2;
  int o = (int)(t % O);
  int b = (int)(t / O);
  float step = 2.f * PIf * (float)y / (float)n2;
  float s0, c0, s1, c1;
  sincosf(-step * (float)(L - 1), &s0, &c0);
  sincosf(step, &s1, &c1);
  float2 w = make_float2(c0, s0), e = make_float2(c1, s1);
  float acc = 0.f;
  for (int m = 0; m < P; ++m) {
    float2 g = G1[(long long)(((b * n2 + kk) * P + m) * O + o) * n2 + x];
    acc += g.x * w.x - g.y * w.y;
    w = cmul(w, e);
  }
  out[(long long)(((b * O + o) * n2 + kk) * n2 + y) * n2 + x] = acc;
}

// ---------------- BN (relu -> normalize per channel) ----------------
__global__ void k_bn_stats(const float* __restrict__ h, int O, int perbo,
                           float* __restrict__ mu, float* __restrict__ rsig) {
  const int B = 16;
  int o = blockIdx.x;
  long long perch = (long long)B * perbo;
  double s = 0.0, s2 = 0.0;
  for (long long j = threadIdx.x; j < perch; j += blockDim.x) {
    int b = (int)(j / perbo);
    long long r = j % perbo;
    float v = h[((long long)b * O + o) * perbo + r];
    v = v > 0.f ? v : 0.f;
    s += v;
    s2 += (double)v * v;
  }
  __shared__ double sh1[256], sh2[256];
  sh1[threadIdx.x] = s;
  sh2[threadIdx.x] = s2;
  __syncthreads();
  for (int st = 128; st > 0; st >>= 1) {
    if ((int)threadIdx.x < st) {
      sh1[threadIdx.x] += sh1[threadIdx.x + st];
      sh2[threadIdx.x] += sh2[threadIdx.x + st];
    }
    __syncthreads();
  }
  if (threadIdx.x == 0) {
    double mean = sh1[0] / (double)perch;
    double var = sh2[0] / (double)perch - mean * mean;
    mu[o] = (float)mean;
    rsig[o] = (float)(1.0 / sqrt(var + 1e-5));
  }
}

__global__ void k_bn_apply(float* __restrict__ h, int O, int perbo,
                           const float* __restrict__ mu,
                           const float* __restrict__ rsig, long long total) {
  long long idx = (long long)blockIdx.x * blockDim.x + threadIdx.x;
  if (idx >= total) return;
  int o = (int)((idx / perbo) % O);
  float v = h[idx];
  v = v > 0.f ? v : 0.f;
  h[idx] = (v - mu[o]) * rsig[o];
}

// ---------------- SO(3) layer pieces ----------------
// pass 1: DFT over last axis.  xf1[b][i][k][y][n]
__global__ void k_dft_x(const float* __restrict__ h, int I, int nin, int P,
                        int L, float2* __restrict__ xf1) {
  const int B = 16;
  long long total = (long long)B * I * nin * nin * P;
  long long idx = (long long)blockIdx.x * blockDim.x + threadIdx.x;
  if (idx >= total) return;
  int n = (int)(idx % P);
  long long t = idx / P;
  int y = (int)(t % nin); t /= nin;
  int k = (int)(t % nin); t /= nin;
  int i = (int)(t % I);
  int b = (int)(t / I);
  float fn = (float)(n - (L - 1));
  const float* base = h + ((((long long)b * I + i) * nin + k) * nin + y) * nin;
  float es, ec;
  sincosf(-2.f * PIf * fn / (float)nin, &es, &ec);
  float2 e = make_float2(ec, es), w = make_float2(1.f, 0.f);
  float2 acc = make_float2(0.f, 0.f);
  for (int xx = 0; xx < nin; ++xx) {
    acc.x += base[xx] * w.x;
    acc.y += base[xx] * w.y;
    w = cmul(w, e);
  }
  xf1[((((long long)b * I + i) * nin + k) * nin + y) * P + n] = acc;
}

// pass 2: DFT over y.  xf2[b][i][k][m][n]
__global__ void k_dft_y(const float2* __restrict__ xf1, int I, int nin, int P,
                        int L, float2* __restrict__ xf2) {
  const int B = 16;
  long long total = (long long)B * I * nin * P * P;
  long long idx = (long long)blockIdx.x * blockDim.x + threadIdx.x;
  if (idx >= total) return;
  int n = (int)(idx % P);
  long long t = idx / P;
  int m = (int)(t % P); t /= P;
  int k = (int)(t % nin); t /= nin;
  int i = (int)(t % I);
  int b = (int)(t / I);
  float fm = (float)(m - (L - 1));
  float es, ec;
  sincosf(-2.f * PIf * fm / (float)nin, &es, &ec);
  float2 e = make_float2(ec, es), w = make_float2(1.f, 0.f);
  float2 acc = make_float2(0.f, 0.f);
  for (int y = 0; y < nin; ++y) {
    float2 v = xf1[((((long long)b * I + i) * nin + k) * nin + y) * P + n];
    acc.x += v.x * w.x - v.y * w.y;
    acc.y += v.x * w.y + v.y * w.x;
    w = cmul(w, e);
  }
  xf2[((((long long)b * I + i) * nin + k) * P + m) * P + n] = acc;
}

// X[l][(b,m)][(i,n)] = sum_k din[l][m][n][k]*wq[k]*xf2[b][i][k][m][n]
__global__ void k_so3_X(const float2* __restrict__ xf2,
                        const float* __restrict__ din,
                        const float* __restrict__ wq, int L, int P, int I,
                        int nin, float2* __restrict__ X) {
  const int B = 16;
  long long total = (long long)L * B * P * I * P;
  long long idx = (long long)blockIdx.x * blockDim.x + threadIdx.x;
  if (idx >= total) return;
  int n = (int)(idx % P);
  long long t = idx / P;
  int i = (int)(t % I); t /= I;
  int m = (int)(t % P); t /= P;
  int b = (int)(t % B);
  int l = (int)(t / B);
  float2 acc = make_float2(0.f, 0.f);
  for (int k = 0; k < nin; ++k) {
    float w = din[((l * P + m) * P + n) * nin + k] * wq[k];
    float2 v = xf2[((((long long)b * I + i) * nin + k) * P + m) * P + n];
    acc.x += w * v.x;
    acc.y += w * v.y;
  }
  X[(long long)((l * B + b) * P + m) * (I * P) + i * P + n] = acc;
}

// Y[l][(i,m)][(n,o)] = sum_p wc[l][m][n][p] * kern[i][o][p]
__global__ void k_so3_Y(const float2* __restrict__ wc,
                        const float* __restrict__ kern, int L, int P, int I,
                        int O, float2* __restrict__ Y) {
  long long total = (long long)L * I * P * P * O;
  long long idx = (long long)blockIdx.x * blockDim.x + threadIdx.x;
  if (idx >= total) return;
  int o = (int)(idx % O);
  long long t = idx / O;
  int n = (int)(t % P); t /= P;
  int m = (int)(t % P); t /= P;
  int i = (int)(t % I);
  int l = (int)(t / I);
  const float2* w = wc + ((long long)(l * P + m) * P + n) * 144;
  const float* kr = kern + ((long long)i * O + o) * 144;
  float2 acc = make_float2(0.f, 0.f);
  for (int p = 0; p < 144; ++p) {
    float kv = kr[p];
    acc.x += w[p].x * kv;
    acc.y += w[p].y * kv;
  }
  Y[(long long)((l * I + i) * P + m) * (P * O) + n * O + o] = acc;
}

// ---------------- integrate + head ----------------
__global__ void k_integrate(const float* __restrict__ h,
                            const float* __restrict__ wq5,
                            float* __restrict__ feats) {
  int idx = blockIdx.x * blockDim.x + threadIdx.x;
  if (idx >= 16 * 256) return;
  int c = idx % 256;
  int b = idx / 256;
  float acc = 0.f;
  for (int k = 0; k < 8; ++k) {
    const float* base = h + ((((long long)b * 256 + c) * 8 + k) * 64);
    float s = 0.f;
    for (int j = 0; j < 64; ++j) s += base[j];
    acc += (s / 64.f) * wq5[k];
  }
  feats[idx] = acc;
}

__global__ void k_head(const float* __restrict__ feats,
                       const float* __restrict__ wgt,
                       const float* __restrict__ fc1w,
                       const float* __restrict__ fc1b,
                       const float* __restrict__ fc2w,
                       const float* __restrict__ fc2b, float* __restrict__ out) {
  __shared__ float z[16 * 256];
  __shared__ float hd[16 * 128];
  int tid = threadIdx.x;
  for (int idx = tid; idx < 16 * 256; idx += blockDim.x) {
    int c = idx % 256, b = idx / 256;
    float v = 0.f;
    for (int g = 0; g < 4; ++g) v += feats[(g * 16 + b) * 256 + c] * wgt[c * 4 + g];
    z[idx] = v;
  }
  __syncthreads();
  for (int idx = tid; idx < 16 * 128; idx += blockDim.x) {
    int j = idx % 128, b = idx / 128;
    float v = fc1b[j];
    for (int c = 0; c < 256; ++c) v += z[b * 256 + c] * fc1w[c * 128 + j];
    hd[idx] = v > 0.f ? v : 0.f;
  }
  __syncthreads();
  for (int idx = tid; idx < 160; idx += blockDim.x) {
    int tc = idx % 10, b = idx / 10;
    float v = fc2b[tc];
    for (int j = 0; j < 128; ++j) v += hd[b * 128 + j] * fc2w[j * 10 + tc];
    out[idx] = v;
  }
}

// ---------------- host orchestration ----------------
static inline unsigned gblk(long long total) {
  return (unsigned)((total + 255) / 256);
}

extern "C" void kernel_launch(void* const* d_in, const int* in_sizes, int n_in,
                              void* d_out, int out_size, void* d_ws,
                              size_t ws_size, hipStream_t stream) {
  (void)in_sizes; (void)n_in; (void)out_size;
  const float* x = (const float*)d_in[0];
  const float* kconv[5] = {(const float*)d_in[1], (const float*)d_in[2],
                           (const float*)d_in[3], (const float*)d_in[4],
                           (const float*)d_in[5]};
  const float* wgt = (const float*)d_in[6];
  const float* fc1w = (const float*)d_in[7];
  const float* fc1b = (const float*)d_in[8];
  const float* fc2w = (const float*)d_in[9];
  const float* fc2b = (const float*)d_in[10];
  float* out = (float*)d_out;

  const int BWh[6] = {32, 16, 10, 8, 6, 4};
  const int FEATh[6] = {1, 16, 32, 64, 128, 256};
  const int qoff[6] = {0, 64, 96, 116, 132, 144};

  // ---- workspace arena ----
  uintptr_t base = (uintptr_t)d_ws;
  size_t off = 0;
  auto alloc = [&](size_t nbytes) -> void* {
    off = (off + 255) & ~(size_t)255;
    void* p = (void*)(base + off);
    off += nbytes;
    return p;
  };
  float* sc = (float*)alloc(1024 * sizeof(float));
  float* din[5]; float* dou[5]; float* dkt[5];
  for (int ci = 0; ci < 5; ++ci) {
    int L = BWh[ci + 1], P = 2 * L - 1;
    din[ci] = (float*)alloc((size_t)L * P * P * (2 * BWh[ci]) * 4);
    dou[ci] = (float*)alloc((size_t)L * P * P * (2 * BWh[ci + 1]) * 4);
    dkt[ci] = (float*)alloc((size_t)L * P * P * (ci == 0 ? 24 : 144) * 4);
  }
  float2* wcs2 = (float2*)alloc((size_t)16 * 31 * 24 * 8);
  float2* wcl[4];
  for (int ci = 1; ci < 5; ++ci) {
    int L = BWh[ci + 1], P = 2 * L - 1;
    wcl[ci - 1] = (float2*)alloc((size_t)L * P * P * 144 * 8);
  }
  float* hA = (float*)alloc((size_t)8388608 * 4);
  float* hB = (float*)alloc((size_t)8388608 * 4);
  float2* regA = (float2*)alloc((size_t)7872512 * 8);  // xf1 | Fm
  float2* regB = (float2*)alloc((size_t)8126464 * 8);  // xf/xf2 | G1
  float2* Xb = (float2*)alloc((size_t)924160 * 8);
  float2* Yb = (float2*)alloc((size_t)6422528 * 8);
  float2* Zb = (float2*)alloc((size_t)3936256 * 8);
  float* mu = (float*)alloc(256 * 4);
  float* rs = (float*)alloc(256 * 4);
  float* feats = (float*)alloc((size_t)4 * 16 * 256 * 4);
  if (off > ws_size) return;  // workspace too small: bail deterministically

  // ---- constant tables (deterministic; rebuilt every launch) ----
  k_smallconst<<<1, 256, 0, stream>>>(sc);
  for (int ci = 0; ci < 5; ++ci) {
    int L = BWh[ci + 1], P = 2 * L - 1;
    long long t1 = (long long)L * P * P * (2 * BWh[ci]);
    k_wigner<<<gblk(t1), 256, 0, stream>>>(L, 2 * BWh[ci], sc + qoff[ci], din[ci]);
    long long t2 = (long long)L * P * P * (2 * BWh[ci + 1]);
    k_wigner<<<gblk(t2), 256, 0, stream>>>(L, 2 * BWh[ci + 1], sc + qoff[ci + 1], dou[ci]);
    int nbk = (ci == 0) ? 24 : 144;
    const float* bet = (ci == 0) ? (sc + 344) : (sc + 512);
    long long t3 = (long long)L * P * P * nbk;
    k_wigner<<<gblk(t3), 256, 0, stream>>>(L, nbk, bet, dkt[ci]);
  }
  k_wc_s2<<<gblk(16 * 31 * 24), 256, 0, stream>>>(dkt[0], sc + 320, wcs2);
  for (int ci = 1; ci < 5; ++ci) {
    int L = BWh[ci + 1], P = 2 * L - 1;
    k_wc_so3<<<gblk((long long)L * P * P * 144), 256, 0, stream>>>(
        L, dkt[ci], sc + 368, sc + 656, wcl[ci - 1]);
  }

  // ---- per-group pipeline ----
  for (int g = 0; g < 4; ++g) {
    const float* xg = x + (size_t)g * 16 * 64 * 64;
    {  // S2 conv: b 32 -> 16, feat 1 -> 16
      const int L = 16, P = 31, O = 16, n2 = 32;
      float2* xf = regB;
      k_s2_dft<<<gblk(16 * 64 * 31), 256, 0, stream>>>(xg, xf);
      k_s2_X<<<gblk(16 * 16 * 31), 256, 0, stream>>>(din[0], sc + 160, xf, Xb);
      k_s2_Y<<<gblk(16 * 31 * 16), 256, 0, stream>>>(wcs2, kconv[0], Yb);
      k_s2_Z<<<gblk(16LL * 16 * 31 * 31 * 16), 256, 0, stream>>>(Xb, Yb, Zb);
      k_fm<<<gblk((long long)16 * n2 * P * P * O), 256, 0, stream>>>(
          Zb, dou[0], L, P, O, n2, regA);
      k_idft1<<<gblk((long long)16 * n2 * P * O * n2), 256, 0, stream>>>(
          regA, L, P, O, n2, regB);
      k_idft2<<<gblk((long long)16 * O * n2 * n2 * n2), 256, 0, stream>>>(
          regB, L, P, O, n2, hA);
      int perbo = n2 * n2 * n2;
      k_bn_stats<<<O, 256, 0, stream>>>(hA, O, perbo, mu, rs);
      k_bn_apply<<<gblk((long long)16 * O * perbo), 256, 0, stream>>>(
          hA, O, perbo, mu, rs, (long long)16 * O * perbo);
    }
    float* hcur = hA;
    float* hnxt = hB;
    for (int ci = 1; ci < 5; ++ci) {  // SO(3) convs
      int L = BWh[ci + 1], nin = 2 * BWh[ci], P = 2 * L - 1, n2 = 2 * L;
      int I = FEATh[ci], O = FEATh[ci + 1];
      k_dft_x<<<gblk((long long)16 * I * nin * nin * P), 256, 0, stream>>>(
          hcur, I, nin, P, L, regA);
      k_dft_y<<<gblk((long long)16 * I * nin * P * P), 256, 0, stream>>>(
          regA, I, nin, P, L, regB);
      k_so3_X<<<gblk((long long)L * 16 * P * I * P), 256, 0, stream>>>(
          regB, din[ci], sc + 160 + qoff[ci], L, P, I, nin, Xb);
      k_so3_Y<<<gblk((long long)L * I * P * P * O), 256, 0, stream>>>(
          wcl[ci - 1], kconv[ci], L, P, I, O, Yb);
      int M = 16 * P, N = P * O, Kk = I * P;
      int tilesN = N / 16, tilesM = M / 16;
      dim3 gg((unsigned)((tilesN + 7) / 8), (unsigned)tilesM, (unsigned)L);
      dim3 bb(32, 8);
      cgemm16_wmma<<<gg, bb, 0, stream>>>(Xb, Yb, Zb, M, N, Kk,
                                          (long long)M * Kk, (long long)Kk * N,
                                          (long long)M * N);
      k_fm<<<gblk((long long)16 * n2 * P * P * O), 256, 0, stream>>>(
          Zb, dou[ci], L, P, O, n2, regA);
      k_idft1<<<gblk((long long)16 * n2 * P * O * n2), 256, 0, stream>>>(
          regA, L, P, O, n2, regB);
      k_idft2<<<gblk((long long)16 * O * n2 * n2 * n2), 256, 0, stream>>>(
          regB, L, P, O, n2, hnxt);
      int perbo = n2 * n2 * n2;
      k_bn_stats<<<O, 256, 0, stream>>>(hnxt, O, perbo, mu, rs);
      k_bn_apply<<<gblk((long long)16 * O * perbo), 256, 0, stream>>>(
          hnxt, O, perbo, mu, rs, (long long)16 * O * perbo);
      float* tmp = hcur; hcur = hnxt; hnxt = tmp;
    }
    k_integrate<<<gblk(16 * 256), 256, 0, stream>>>(hcur, sc + 160 + 144,
                                                    feats + g * 16 * 256);
  }
  k_head<<<1, 256, 0, stream>>>(feats, wgt, fc1w, fc1b, fc2w, fc2b, out);
}